// SlidingWindowAttention_88897233093064
// MI455X (gfx1250) — compile-verified
//
#include <hip/hip_runtime.h>
#include <math.h>

// ---------------------------------------------------------------------------
// Sliding-window MHA for MI455X (gfx1250, wave32, WMMA).
// All four GEMMs (QKV proj, scores, ctx, out-proj) use V_WMMA_F32_16X16X4_F32.
// ---------------------------------------------------------------------------

typedef __attribute__((ext_vector_type(2))) float v2f;
typedef __attribute__((ext_vector_type(8))) float v8f;

// D = A(16x4) * B(4x16) + C(16x16), all fp32.
// 8-arg form: (neg_a, A, neg_b, B, c_mod, C, reuse_a, reuse_b)
#define WMMA_F32X4(a, b, c) \
  __builtin_amdgcn_wmma_f32_16x16x4_f32(false, (a), false, (b), (short)0, (c), false, false)

constexpr int S   = 512;   // sequence length (rows)
constexpr int L   = 2048;  // raw signal length
constexpr int NW  = 63;    // number of windows
constexpr int E   = 64;    // embed dim == window size
constexpr int H   = 8;     // heads
constexpr int HD  = 8;     // head dim
constexpr int STR = 32;    // window stride
constexpr int ST  = 16;    // query rows per tile (WMMA M)
constexpr int STILES = S / ST;                 // 32
constexpr size_t PROJ_ELEMS = (size_t)NW * H * S * HD;   // 2,064,384 per tensor
constexpr size_t X_SIZE     = (size_t)S * NW * E;        // x output elems

// ===========================================================================
// Kernel 1: packed QKV projection.
// Block = (window nw, s-tile). Computes qp/kp/vp [16 x 64] tiles via WMMA,
// stores head-major [NW, H, S, HD] into workspace.
// ===========================================================================
__global__ __launch_bounds__(128)
void swa_proj_kernel(const float* __restrict__ q, const float* __restrict__ k,
                     const float* __restrict__ v, const float* __restrict__ W,
                     const float* __restrict__ bias, float* __restrict__ ws) {
  __shared__ float At[3][ST][E];   // 12 KB: unfolded q/k/v window tiles

  const int blk  = blockIdx.x;
  const int nw   = blk / STILES;
  const int s0   = (blk % STILES) * ST;
  const int tid  = threadIdx.x;

  const float* src[3] = {q, k, v};
  #pragma unroll
  for (int t = 0; t < 3; ++t)
    for (int idx = tid; idx < ST * E; idx += 128) {
      int r = idx >> 6, c = idx & 63;
      At[t][r][c] = src[t][(size_t)(s0 + r) * L + nw * STR + c];
    }
  __syncthreads();

  const int lane = tid & 31;
  const int et   = tid >> 5;                 // e-tile 0..3 (one per wave)
  const int m    = lane & 15;                // row / col within tile
  const int kOff = (lane < 16) ? 0 : 2;      // A/B frag K sub-offset
  const int rOff = (lane < 16) ? 0 : 8;      // C row offset

  #pragma unroll
  for (int sel = 0; sel < 3; ++sel) {        // 0=Q, 1=K, 2=V
    v8f c = {};
    #pragma unroll
    for (int k0 = 0; k0 < E; k0 += 4) {
      v2f a, b;
      a.x = At[sel][m][k0 + kOff];
      a.y = At[sel][m][k0 + kOff + 1];
      // B[k][n] = W[sel*64 + et*16 + n][k0+k]   (C = A * W^T)
      const float* wr = W + (size_t)(sel * E + et * 16 + m) * E + k0 + kOff;
      b.x = wr[0];
      b.y = wr[1];
      c = WMMA_F32X4(a, b, c);
    }
    const float bv  = bias[sel * E + et * 16 + m];
    float* dst      = ws + (size_t)sel * PROJ_ELEMS;
    const int e     = et * 16 + m;
    const int h     = e >> 3;
    const int hd    = e & 7;
    #pragma unroll
    for (int j = 0; j < 8; ++j) {
      int srow = s0 + j + rOff;
      dst[((size_t)(nw * H + h) * S + srow) * HD + hd] = c[j] + bv;
    }
  }
}

// ===========================================================================
// Kernel 2: attention + out-projection + head-mean.
// Block = (window nw, 16-query tile). Loops 8 heads sequentially; head-mean
// of attn accumulates in registers (64 per thread == the 16x512 tile).
// ===========================================================================
__global__ __launch_bounds__(128)
void swa_attn_kernel(const float* __restrict__ ws, const float* __restrict__ Wo,
                     const float* __restrict__ bo, float* __restrict__ xout,
                     float* __restrict__ attn_out) {
  __shared__ float sc[ST][S];       // 32 KB: scores/attn for current head
  __shared__ float vsh[S][HD];      // 16 KB: V for current head
  __shared__ float ctxAcc[ST][E];   //  4 KB: merged-head context tile

  const int blk  = blockIdx.x;
  const int nw   = blk / STILES;
  const int s0   = (blk % STILES) * ST;
  const int tid  = threadIdx.x;
  const int lane = tid & 31;
  const int wave = tid >> 5;
  const int m    = lane & 15;
  const int kOff = (lane < 16) ? 0 : 2;
  const int rOff = (lane < 16) ? 0 : 8;
  const int r    = tid >> 3;        // softmax: row 0..15
  const int i    = tid & 7;         // softmax: 8 lanes per row

  const float* qp = ws;
  const float* kp = ws + PROJ_ELEMS;
  const float* vp = ws + 2 * PROJ_ELEMS;

  for (int idx = tid; idx < ST * E; idx += 128)
    (&ctxAcc[0][0])[idx] = 0.f;

  float macc[64];                    // running head-mean, [r][i + 8j]
  #pragma unroll
  for (int j = 0; j < 64; ++j) macc[j] = 0.f;

  const float rscale = 0.35355339059327373f;   // 1/sqrt(HD)

  for (int h = 0; h < H; ++h) {
    const float* qh = qp + (size_t)(nw * H + h) * S * HD;
    const float* kh = kp + (size_t)(nw * H + h) * S * HD;
    const float* vh = vp + (size_t)(nw * H + h) * S * HD;

    // stage V[512][8] into LDS (coalesced)
    for (int idx = tid; idx < S * HD; idx += 128)
      (&vsh[0][0])[idx] = vh[idx];

    // Q fragments for the two K=4 steps (K total = 8)
    v2f a0, a1;
    {
      const float* qr = qh + (size_t)(s0 + m) * HD;
      a0.x = qr[kOff];     a0.y = qr[kOff + 1];
      a1.x = qr[4 + kOff]; a1.y = qr[4 + kOff + 1];
    }

    // --- scores: S[16,512] = Qh * Kh^T * scale, wave-strided key tiles ---
    for (int kt = wave; kt < S / 16; kt += 4) {
      const float* kr = kh + (size_t)(kt * 16 + m) * HD;
      v2f b0, b1;
      b0.x = kr[kOff];     b0.y = kr[kOff + 1];
      b1.x = kr[4 + kOff]; b1.y = kr[4 + kOff + 1];
      v8f c = {};
      c = WMMA_F32X4(a0, b0, c);
      c = WMMA_F32X4(a1, b1, c);
      #pragma unroll
      for (int j = 0; j < 8; ++j)
        sc[j + rOff][kt * 16 + m] = c[j] * rscale;
    }
    __syncthreads();

    // --- softmax over each row of 512 (8 lanes/row, shfl-xor reductions) ---
    float mx = -INFINITY;
    #pragma unroll
    for (int j = 0; j < 64; ++j) mx = fmaxf(mx, sc[r][i + 8 * j]);
    mx = fmaxf(mx, __shfl_xor(mx, 1, 8));
    mx = fmaxf(mx, __shfl_xor(mx, 2, 8));
    mx = fmaxf(mx, __shfl_xor(mx, 4, 8));
    float sum = 0.f;
    #pragma unroll
    for (int j = 0; j < 64; ++j) {
      float ex = __expf(sc[r][i + 8 * j] - mx);
      sc[r][i + 8 * j] = ex;
      sum += ex;
    }
    sum += __shfl_xor(sum, 1, 8);
    sum += __shfl_xor(sum, 2, 8);
    sum += __shfl_xor(sum, 4, 8);
    const float inv = 1.0f / sum;
    #pragma unroll
    for (int j = 0; j < 64; ++j) {
      float aw = sc[r][i + 8 * j] * inv;
      sc[r][i + 8 * j] = aw;
      macc[j] += aw * 0.125f;      // head-mean accumulation in registers
    }
    __syncthreads();

    // --- ctx[16,8] = attn[16,512] @ V[512,8]; K split across 4 waves ---
    v8f cc = {};
    const int kBase = wave * (S / 4);      // 128 keys per wave
    #pragma unroll
    for (int k0 = 0; k0 < S / 4; k0 += 4) {
      const int kk = kBase + k0;
      v2f a, b;
      a.x = sc[m][kk + kOff];
      a.y = sc[m][kk + kOff + 1];
      b.x = (m < HD) ? vsh[kk + kOff][m] : 0.f;       // B[k][n] = V[k][n], n<8
      b.y = (m < HD) ? vsh[kk + kOff + 1][m] : 0.f;
      cc = WMMA_F32X4(a, b, cc);
    }
    if (m < HD) {
      #pragma unroll
      for (int j = 0; j < 8; ++j)
        atomicAdd(&ctxAcc[j + rOff][h * HD + m], cc[j]);   // ds_add_f32
    }
    __syncthreads();
  }

  // --- out projection: x[16,64] = ctx[16,64] @ Wo^T + bo (wave owns e-tile) ---
  {
    v8f co = {};
    #pragma unroll
    for (int k0 = 0; k0 < E; k0 += 4) {
      v2f a, b;
      a.x = ctxAcc[m][k0 + kOff];
      a.y = ctxAcc[m][k0 + kOff + 1];
      const float* wr = Wo + (size_t)(wave * 16 + m) * E + k0 + kOff;
      b.x = wr[0];
      b.y = wr[1];
      co = WMMA_F32X4(a, b, co);
    }
    const float bv = bo[wave * 16 + m];
    #pragma unroll
    for (int j = 0; j < 8; ++j) {
      int s = s0 + j + rOff;
      xout[((size_t)s * NW + nw) * E + wave * 16 + m] = co[j] + bv;
    }
  }

  // --- write head-averaged attention weights [NW, S, S] ---
  #pragma unroll
  for (int j = 0; j < 64; ++j) {
    size_t t = (size_t)i + 8 * j;
    attn_out[((size_t)nw * S + (s0 + r)) * S + t] = macc[j];
  }
}

// ===========================================================================
extern "C" void kernel_launch(void* const* d_in, const int* in_sizes, int n_in,
                              void* d_out, int out_size, void* d_ws, size_t ws_size,
                              hipStream_t stream) {
  const float* q  = (const float*)d_in[0];
  const float* k  = (const float*)d_in[1];
  const float* v  = (const float*)d_in[2];
  const float* W  = (const float*)d_in[3];   // in_proj_weight [192,64]
  const float* b  = (const float*)d_in[4];   // in_proj_bias [192]
  const float* Wo = (const float*)d_in[5];   // out_proj_weight [64,64]
  const float* bo = (const float*)d_in[6];   // out_proj_bias [64]

  float* ws       = (float*)d_ws;            // qp|kp|vp, 3 * 8.25 MB
  float* xout     = (float*)d_out;           // [S, NW, E]
  float* attn_out = (float*)d_out + X_SIZE;  // [NW, S, S]

  const dim3 grid(NW * STILES);              // 2016 blocks
  const dim3 block(128);                     // 4 waves

  swa_proj_kernel<<<grid, block, 0, stream>>>(q, k, v, W, b, ws);
  swa_attn_kernel<<<grid, block, 0, stream>>>(ws, Wo, bo, xout, attn_out);
}